// SoftKNN_82377472737431
// MI455X (gfx1250) — compile-verified
//
#include <hip/hip_runtime.h>
#include <math.h>

typedef __attribute__((ext_vector_type(2))) float v2f;
typedef __attribute__((ext_vector_type(8))) float v8f;

#define NPTS 8192
#define KDIM 128
#define TOPK 6          // MIN_K + 1
#define MINK 5
#define SPLIT 8
#define TILES_PER_SPLIT (NPTS / 16 / SPLIT)   // 64 column tiles per split
#define EPS_SHIFT 1e-10f
#define EPS_NORM  1e-12f
#define FINF 3.0e38f

// ---------------- 1) row squared norms ----------------
__global__ void sq_kernel(const float* __restrict__ f, float* __restrict__ sq) {
    int row = blockIdx.x * blockDim.x + threadIdx.x;
    if (row >= NPTS) return;
    const float4* p = (const float4*)(f + (size_t)row * KDIM);
    float s = 0.f;
#pragma unroll 4
    for (int i = 0; i < KDIM / 4; ++i) {
        float4 v = p[i];
        s += v.x * v.x + v.y * v.y + v.z * v.z + v.w * v.w;
    }
    sq[row] = s;
}

// ---------------- 2) per-(rowblock, split) top-6 via WMMA ----------------
__global__ __launch_bounds__(32) void topk_kernel(const float* __restrict__ f,
                                                  const float* __restrict__ sq,
                                                  float* __restrict__ cand) {
    const int mb    = blockIdx.x * 16;     // row-block base
    const int split = blockIdx.y;
    const int lane  = threadIdx.x & 31;
    const int lo    = lane & 15;
    const int hi    = lane >> 4;

    // A-tile resident: 16x128 fp32 as 32 chunks of 16x4
    v2f a[KDIM / 4];
    const float* Ap = f + (size_t)(mb + lo) * KDIM + 2 * hi;
#pragma unroll
    for (int c = 0; c < KDIM / 4; ++c) a[c] = *(const v2f*)(Ap + 4 * c);

    float sqm[8];
#pragma unroll
    for (int r = 0; r < 8; ++r) sqm[r] = sq[mb + r + 8 * hi];

    // per-lane sorted top-6 for each of the 8 rows this lane covers
    float t[8][TOPK];
#pragma unroll
    for (int r = 0; r < 8; ++r)
#pragma unroll
        for (int j = 0; j < TOPK; ++j) t[r][j] = FINF;

    for (int nt = 0; nt < TILES_PER_SPLIT; ++nt) {
        const int nb = (split * TILES_PER_SPLIT + nt) * 16;
        const float* Bp = f + (size_t)(nb + lo) * KDIM + 2 * hi;
        v8f acc = {};
#pragma unroll
        for (int c = 0; c < KDIM / 4; ++c) {
            v2f b = *(const v2f*)(Bp + 4 * c);
            acc = __builtin_amdgcn_wmma_f32_16x16x4_f32(
                false, a[c], false, b, (short)0, acc, false, false);
        }
        const float sqn = sq[nb + lo];
#pragma unroll
        for (int r = 0; r < 8; ++r) {
            float d2 = sqm[r] + sqn - 2.0f * acc[r];
            d2 = fmaxf(d2, 1e-30f);
            float d = sqrtf(d2);
            if (d < t[r][TOPK - 1]) {
                t[r][TOPK - 1] = d;
#pragma unroll
                for (int j = TOPK - 1; j > 0; --j) {
                    float a0 = fminf(t[r][j - 1], t[r][j]);
                    float a1 = fmaxf(t[r][j - 1], t[r][j]);
                    t[r][j - 1] = a0;
                    t[r][j]     = a1;
                }
            }
        }
    }

    // merge the 16 per-lane sorted lists of each row (butterfly min-extraction)
#pragma unroll
    for (int r = 0; r < 8; ++r) {
        int idx = 0;
        float mins[TOPK];
        for (int it = 0; it < TOPK; ++it) {
            float h = (idx < TOPK) ? t[r][idx] : FINF;
            float m = h;
#pragma unroll
            for (int off = 1; off < 16; off <<= 1)
                m = fminf(m, __shfl_xor(m, off, 16));
            if (h == m && idx < TOPK) idx++;
            mins[it] = m;
        }
        if (lo == 0) {
            const int row = mb + r + 8 * hi;
            float* cp = cand + ((size_t)row * SPLIT + split) * TOPK;
#pragma unroll
            for (int it = 0; it < TOPK; ++it) cp[it] = mins[it];
        }
    }
}

// ---------------- 3) merge splits -> threshold (6th smallest) ----------------
__global__ void merge_kernel(const float* __restrict__ cand, float* __restrict__ thr) {
    int row = blockIdx.x * blockDim.x + threadIdx.x;
    if (row >= NPTS) return;
    float best[TOPK];
#pragma unroll
    for (int j = 0; j < TOPK; ++j) best[j] = FINF;
    const float* cp = cand + (size_t)row * SPLIT * TOPK;
    for (int i = 0; i < SPLIT * TOPK; ++i) {
        float d = cp[i];
        if (d < best[TOPK - 1]) {
            best[TOPK - 1] = d;
#pragma unroll
            for (int j = TOPK - 1; j > 0; --j) {
                float a0 = fminf(best[j - 1], best[j]);
                float a1 = fmaxf(best[j - 1], best[j]);
                best[j - 1] = a0;
                best[j]     = a1;
            }
        }
    }
    thr[row] = best[MINK];
}

// ---------------- 4) unnormalized weights + partial row sums (WMMA) ----------------
__global__ __launch_bounds__(32) void weights_kernel(const float* __restrict__ f,
                                                     const float* __restrict__ sq,
                                                     const float* __restrict__ thr,
                                                     float* __restrict__ out,
                                                     float* __restrict__ psum) {
    const int mb    = blockIdx.x * 16;
    const int split = blockIdx.y;
    const int lane  = threadIdx.x & 31;
    const int lo    = lane & 15;
    const int hi    = lane >> 4;

    v2f a[KDIM / 4];
    const float* Ap = f + (size_t)(mb + lo) * KDIM + 2 * hi;
#pragma unroll
    for (int c = 0; c < KDIM / 4; ++c) a[c] = *(const v2f*)(Ap + 4 * c);

    float sqm[8], thrm[8], rs[8];
#pragma unroll
    for (int r = 0; r < 8; ++r) {
        sqm[r]  = sq[mb + r + 8 * hi];
        thrm[r] = thr[mb + r + 8 * hi];
        rs[r]   = 0.0f;
    }

    for (int nt = 0; nt < TILES_PER_SPLIT; ++nt) {
        const int nb = (split * TILES_PER_SPLIT + nt) * 16;
        const float* Bp = f + (size_t)(nb + lo) * KDIM + 2 * hi;
        v8f acc = {};
#pragma unroll
        for (int c = 0; c < KDIM / 4; ++c) {
            v2f b = *(const v2f*)(Bp + 4 * c);
            acc = __builtin_amdgcn_wmma_f32_16x16x4_f32(
                false, a[c], false, b, (short)0, acc, false, false);
        }
        const float sqn = sq[nb + lo];
#pragma unroll
        for (int r = 0; r < 8; ++r) {
            float d2 = sqm[r] + sqn - 2.0f * acc[r];
            d2 = fmaxf(d2, 1e-30f);
            float d = sqrtf(d2);
            float w = fmaxf(thrm[r] - d + EPS_SHIFT, 0.0f);
            out[(size_t)(mb + r + 8 * hi) * NPTS + nb + lo] = w;
            rs[r] += w;
        }
    }

    // deterministic butterfly sum across the 16 lanes that share each row
#pragma unroll
    for (int r = 0; r < 8; ++r) {
        float s = rs[r];
#pragma unroll
        for (int off = 1; off < 16; off <<= 1)
            s += __shfl_xor(s, off, 16);
        if (lo == 0) {
            const int row = mb + r + 8 * hi;
            psum[(size_t)row * SPLIT + split] = s;
        }
    }
}

// ---------------- 5) sum partial norms per row ----------------
__global__ void normsum_kernel(const float* __restrict__ psum, float* __restrict__ norm) {
    int row = blockIdx.x * blockDim.x + threadIdx.x;
    if (row >= NPTS) return;
    float s = 0.f;
#pragma unroll
    for (int i = 0; i < SPLIT; ++i) s += psum[(size_t)row * SPLIT + i];
    norm[row] = fmaxf(s, EPS_NORM);
}

// ---------------- 6) in-place row rescale of output ----------------
__global__ void scale_kernel(float* __restrict__ out, const float* __restrict__ norm) {
    size_t i = ((size_t)blockIdx.x * blockDim.x + threadIdx.x) * 4;
    int row = (int)(i >> 13);               // / 8192
    float inv = 1.0f / norm[row];
    float4* p = (float4*)(out + i);
    float4 v = *p;
    v.x *= inv; v.y *= inv; v.z *= inv; v.w *= inv;
    *p = v;
}

extern "C" void kernel_launch(void* const* d_in, const int* in_sizes, int n_in,
                              void* d_out, int out_size, void* d_ws, size_t ws_size,
                              hipStream_t stream) {
    const float* f = (const float*)d_in[0];
    float* out = (float*)d_out;
    float* ws  = (float*)d_ws;

    float* sq   = ws;                                       // 8192
    float* thr  = sq + NPTS;                                // 8192
    float* cand = thr + NPTS;                               // 8192*8*6
    float* psum = cand + (size_t)NPTS * SPLIT * TOPK;       // 8192*8
    float* norm = psum + (size_t)NPTS * SPLIT;              // 8192

    sq_kernel<<<NPTS / 256, 256, 0, stream>>>(f, sq);

    dim3 grid(NPTS / 16, SPLIT);
    topk_kernel<<<grid, 32, 0, stream>>>(f, sq, cand);
    merge_kernel<<<NPTS / 256, 256, 0, stream>>>(cand, thr);
    weights_kernel<<<grid, 32, 0, stream>>>(f, sq, thr, out, psum);
    normsum_kernel<<<NPTS / 256, 256, 0, stream>>>(psum, norm);

    const size_t total_vec4 = (size_t)NPTS * NPTS / 4;      // 16M threads
    scale_kernel<<<(unsigned)(total_vec4 / 256), 256, 0, stream>>>(out, norm);
}